// MSSSIML1Loss_28509992911001
// MI455X (gfx1250) — compile-verified
//
#include <hip/hip_runtime.h>
#include <hip/hip_bf16.h>

// ---------------------------------------------------------------------------
// MS-SSIM + L1 loss, fully fused, CDNA5 (gfx1250) wave32, V_WMMA_F32_16X16X4_F32.
// Separable Gaussian conv as Toeplitz matmul; all 6 conv fields batched through
// shared weight fragments; SSIM combine done in WMMA D-registers (no planes).
// ---------------------------------------------------------------------------

typedef float v2f __attribute__((ext_vector_type(2)));
typedef float v8f __attribute__((ext_vector_type(8)));

#define BATCH   8
#define CHANS   3
#define IMG     512
#define FSZ     33
#define PADR    16
#define NSIG    5

#define TW      64              // output tile width
#define THT     32              // output tile height
#define PW      (TW + 32)       // 96 patch width
#define PH      (THT + 32)      // 64 patch height
#define PSTR    97              // patch LDS row stride (bank-conflict pad)
#define HSTR    65              // H-buffer row stride
#define HROWS   64
#define HBSZ    (HROWS * HSTR)  // 4160 floats per field H-buffer

#define SM_XP   0
#define SM_YP   (SM_XP + PH * PSTR)            // 6208
#define SM_HF   (SM_YP + PH * PSTR)            // +6208
#define SM_WP   (SM_HF + 6 * HBSZ)             // +24960
#define SM_TOT  (SM_WP + 64)                   // 37440 floats
#define SMEM_BYTES (SM_TOT * sizeof(float))    // 149760 B  (< 160KB)

#define C1v     1.0e-4f          // (0.01*DR)^2
#define C2v     9.0e-4f          // (0.03*DR)^2
#define ALPHAv  0.05f

#define WMMA4(A, Bv, C) __builtin_amdgcn_wmma_f32_16x16x4_f32( \
        false, (A), false, (Bv), (short)0, (C), false, false)

// ---------------------------------------------------------------------------
// Prep: recover the 1-D Gaussian taps from each 2-D mask (row sums; exact
// because g2d = outer(g1,g1) with sum(g1)==1).
// ---------------------------------------------------------------------------
__global__ void msssim_prep(const float* __restrict__ gm, float* __restrict__ w1) {
    int t = threadIdx.x;
    if (t < NSIG * FSZ) {
        int s = t / FSZ, i = t - s * FSZ;
        const float* row = gm + (size_t)(3 * s) * (FSZ * FSZ) + (size_t)i * FSZ;
        float acc = 0.f;
        #pragma unroll 1
        for (int j = 0; j < FSZ; ++j) acc += row[j];
        w1[t] = acc;
    }
}

// ---------------------------------------------------------------------------
// One sigma: horizontal conv of NF fields -> LDS H-buffers, vertical conv of
// NF fields -> registers, SSIM combine in registers.
// ---------------------------------------------------------------------------
template <int NF>
__device__ __forceinline__ void process_sigma(
    const float* __restrict__ xp, const float* __restrict__ yp,
    float* __restrict__ Hf, const float* __restrict__ wpad,
    int wv, int n, int hi, float wms,
    float* __restrict__ Pacc, float* __restrict__ Lacc, float* __restrict__ Gacc)
{
    const v8f vzero = {};

    // ---- horizontal pass: H[64][64] per field; 16 tiles, 2 rounds x 8 waves
    #pragma unroll
    for (int r2 = 0; r2 < 2; ++r2) {
        const int tile = wv + r2 * 8;               // 0..15
        const int r0 = (tile >> 2) * 16, c0 = (tile & 3) * 16;
        v8f acc[NF];
        #pragma unroll
        for (int f = 0; f < NF; ++f) acc[f] = vzero;

        #pragma unroll
        for (int j = 0; j < 12; ++j) {
            const int tk = j * 4 + hi * 2;
            const v2f Bv = { wpad[tk - n + 15], wpad[tk + 1 - n + 15] };
            const int o = (r0 + n) * PSTR + c0 + tk;
            const float x0 = xp[o], x1 = xp[o + 1];
            const float y0 = yp[o], y1 = yp[o + 1];
            { v2f A = { x0, x1 };                       acc[0] = WMMA4(A, Bv, acc[0]); }
            { v2f A = { y0, y1 };                       acc[1] = WMMA4(A, Bv, acc[1]); }
            { v2f A = { x0 * x0, x1 * x1 };             acc[2] = WMMA4(A, Bv, acc[2]); }
            { v2f A = { y0 * y0, y1 * y1 };             acc[3] = WMMA4(A, Bv, acc[3]); }
            { v2f A = { x0 * y0, x1 * y1 };             acc[4] = WMMA4(A, Bv, acc[4]); }
            if constexpr (NF == 6) {
              v2f A = { fabsf(x0 - y0), fabsf(x1 - y1) }; acc[5] = WMMA4(A, Bv, acc[5]);
            }
        }
        const int rw = r0 + hi * 8, cw = c0 + n;
        #pragma unroll
        for (int f = 0; f < NF; ++f)
            #pragma unroll
            for (int k = 0; k < 8; ++k)
                Hf[f * HBSZ + (rw + k) * HSTR + cw] = acc[f][k];
    }
    __syncthreads();

    // ---- vertical pass: out[32][64]; 8 tiles, 1 per wave; combine in regs
    {
        const int r0 = (wv >> 2) * 16, c0 = (wv & 3) * 16;
        v8f acc[NF];
        #pragma unroll
        for (int f = 0; f < NF; ++f) acc[f] = vzero;

        #pragma unroll
        for (int j = 0; j < 12; ++j) {
            const int tk = j * 4 + hi * 2;
            const v2f A = { wpad[tk - n + 15], wpad[tk + 1 - n + 15] };
            #pragma unroll
            for (int f = 0; f < NF; ++f) {
                const int ho = f * HBSZ + (r0 + tk) * HSTR + c0 + n;
                v2f Bv = { Hf[ho], Hf[ho + HSTR] };
                acc[f] = WMMA4(A, Bv, acc[f]);
            }
        }
        #pragma unroll
        for (int k = 0; k < 8; ++k) {
            const float mux = acc[0][k], muy = acc[1][k];
            const float mxx = acc[2][k], myy = acc[3][k], mxy = acc[4][k];
            const float sxx = mxx - mux * mux;
            const float syy = myy - muy * muy;
            const float sxy = mxy - mux * muy;
            float cs = (2.f * sxy + C2v) / (sxx + syy + C2v);
            cs = fmaxf(cs, 0.f);
            Pacc[k] *= powf(cs, wms);
            if constexpr (NF == 6) {
                const float l = (2.f * mux * muy + C1v) /
                                (mux * mux + muy * muy + C1v);
                Lacc[k] *= l;
                Gacc[k] += acc[5][k];
            }
        }
    }
}

// ---------------------------------------------------------------------------
// Main fused kernel: one 32x64 output tile per workgroup (256 thr = 8 wave32).
// ---------------------------------------------------------------------------
__global__ void __launch_bounds__(256)
msssim_main(const float* __restrict__ x, const float* __restrict__ y,
            const float* __restrict__ w1, float* __restrict__ partials) {
    extern __shared__ float smem[];
    float* xp   = smem + SM_XP;
    float* yp   = smem + SM_YP;
    float* Hf   = smem + SM_HF;
    float* wpad = smem + SM_WP;

    const int tid  = threadIdx.x;
    const int lane = tid & 31;
    const int wv   = tid >> 5;           // wave 0..7
    const int n    = lane & 15;
    const int hi   = lane >> 4;

    const int blk = blockIdx.x;          // 1024 blocks
    const int b   = blk >> 7;
    const int rem = blk & 127;
    const int oy  = (rem >> 3) * THT;    // 16 tile rows
    const int ox  = (rem & 7) * TW;      // 8 tile cols

    const float msw_tab[NSIG] = {0.0448f, 0.2856f, 0.3001f, 0.2363f, 0.1333f};

    float Pacc[8], Lacc[8], Gacc[8];
    #pragma unroll
    for (int i = 0; i < 8; ++i) { Pacc[i] = 1.f; Lacc[i] = 1.f; Gacc[i] = 0.f; }

    for (int c = 0; c < CHANS; ++c) {
        __syncthreads();   // protect patch overwrite vs previous channel's use
        const float* xg = x + (size_t)(b * CHANS + c) * (IMG * IMG);
        const float* yg = y + (size_t)(b * CHANS + c) * (IMG * IMG);
        for (int idx = tid; idx < PH * PW; idx += 256) {
            int pr = idx / PW, pc = idx - pr * PW;
            int gy = oy + pr - PADR; gy = min(max(gy, 0), IMG - 1);
            int gx = ox + pc - PADR; gx = min(max(gx, 0), IMG - 1);
            xp[pr * PSTR + pc] = xg[(size_t)gy * IMG + gx];
            yp[pr * PSTR + pc] = yg[(size_t)gy * IMG + gx];
        }

        for (int s = 0; s < NSIG; ++s) {
            __syncthreads();
            if (tid < 64) {                   // padded Toeplitz weight table
                int d = tid - 15;
                wpad[tid] = (d >= 0 && d <= 32) ? w1[s * FSZ + d] : 0.f;
            }
            __syncthreads();

            if (s == NSIG - 1)
                process_sigma<6>(xp, yp, Hf, wpad, wv, n, hi, msw_tab[s],
                                 Pacc, Lacc, Gacc);
            else
                process_sigma<5>(xp, yp, Hf, wpad, wv, n, hi, msw_tab[s],
                                 Pacc, Lacc, Gacc);
        }
    }

    // ---- per-workgroup deterministic reduction ----------------------------
    float sum = 0.f;
    #pragma unroll
    for (int i = 0; i < 8; ++i) {
        float ms  = 1.f - Lacc[i] * Pacc[i];
        float gl1 = Gacc[i] * (1.f / 3.f);
        sum += ALPHAv * ms + (1.f - ALPHAv) * gl1;
    }
    float* red = Hf;      // reuse H buffers
    __syncthreads();
    red[tid] = sum;
    __syncthreads();
    for (int off = 128; off > 0; off >>= 1) {
        if (tid < off) red[tid] += red[tid + off];
        __syncthreads();
    }
    if (tid == 0) partials[blk] = red[0];
}

// ---------------------------------------------------------------------------
// Deterministic fixed-order final reduction (no float atomics).
// ---------------------------------------------------------------------------
__global__ void msssim_finalize(const float* __restrict__ partials,
                                float* __restrict__ out, int nparts) {
    if (threadIdx.x == 0 && blockIdx.x == 0) {
        float s = 0.f;
        #pragma unroll 1
        for (int i = 0; i < nparts; ++i) s += partials[i];
        out[0] = s * (1.0f / ((float)BATCH * IMG * IMG));
    }
}

extern "C" void kernel_launch(void* const* d_in, const int* in_sizes, int n_in,
                              void* d_out, int out_size, void* d_ws, size_t ws_size,
                              hipStream_t stream) {
    const float* x  = (const float*)d_in[0];
    const float* yv = (const float*)d_in[1];
    const float* gm = (const float*)d_in[2];
    float* out = (float*)d_out;

    float* w1       = (float*)d_ws;        // 165 floats (5 sigmas x 33 taps)
    float* partials = w1 + 256;            // 1024 floats

    (void)in_sizes; (void)n_in; (void)out_size; (void)ws_size;

    hipFuncSetAttribute(reinterpret_cast<const void*>(msssim_main),
                        hipFuncAttributeMaxDynamicSharedMemorySize,
                        (int)SMEM_BYTES);

    msssim_prep<<<1, 256, 0, stream>>>(gm, w1);
    msssim_main<<<1024, 256, SMEM_BYTES, stream>>>(x, yv, w1, partials);
    msssim_finalize<<<1, 32, 0, stream>>>(partials, out, 1024);
}